// FCCaps_59330678227612
// MI455X (gfx1250) — compile-verified
//
#include <hip/hip_runtime.h>

// FCCaps forward, restructured:
//   U_hat[b,i,k,l] = sum_j U_l[b,i,j] * W[i,k,j,l]          (fp32 WMMA GEMM)
//   S[b,k,l]       = sum_i U_hat[b,i,k,l]
//   A_sum[b,i,k]   = sum_l U_hat[b,i,k,l] * S[b,k,l]        (avoids O(n_l^2) agreement tensor)
//   C = softmax_k(A_sum / sqrt(32)) ; U_h[b,k,l] = sum_i C[b,i,k]*U_hat[b,i,k,l]
//   out = squash(U_h)
//
// Shapes: B=64, n_l=512, n_h=32, d_l=16, d_h=32.
// Workspace: U_hat (64*512*1024 f32 = 128 MiB) + S (64*1024 f32).

typedef __attribute__((ext_vector_type(2))) float v2f;
typedef __attribute__((ext_vector_type(4))) float v4f;
typedef __attribute__((ext_vector_type(8))) float v8f;

#define N_B 64
#define N_L 512
#define N_H 32
#define D_L 16
#define D_H 32
#define KL 1024                       // n_h * d_h
#define UHAT_B_STRIDE ((size_t)N_L * KL)   // 524288 floats per batch

// ---------------------------------------------------------------------------
// Kernel 1: U_hat GEMM via V_WMMA_F32_16X16X4_F32.
// grid = n_l blocks, 256 threads (8 waves).
// Per block (fixed i): (64 x 16) @ (16 x 1024) GEMM.
// wave w: M-tile = (w&3)*16 batches, N half = (w>>2); 32 N-tiles of 16 each.
// ---------------------------------------------------------------------------
__global__ __launch_bounds__(256) void k_uhat_wmma(const float* __restrict__ Ul,
                                                   const float* __restrict__ W,
                                                   float* __restrict__ Uhat) {
  const int i    = blockIdx.x;
  const int wave = threadIdx.x >> 5;
  const int lane = threadIdx.x & 31;
  const int lo16 = lane & 15;
  const int hi   = lane >> 4;      // 0 for lanes 0-15, 1 for lanes 16-31
  const int khi  = hi << 1;        // K stripe offset per ISA A/B layout (0 or 2)
  const int b0   = (wave & 3) << 4;
  const int nh   = wave >> 2;

  // A fragment: U_l[b0+M, i, j]; lane M = lane%16, K = vgpr + 2*(lane/16).
  const float* up = Ul + (size_t)(b0 + lo16) * (N_L * D_L) + (size_t)i * D_L;
  v2f a0 = *(const v2f*)(up + 0  + khi);
  v2f a1 = *(const v2f*)(up + 4  + khi);
  v2f a2 = *(const v2f*)(up + 8  + khi);
  v2f a3 = *(const v2f*)(up + 12 + khi);

  const float* wb = W + (size_t)i * (N_H * D_L * D_H);
  const size_t rowb = (size_t)(b0 + (hi << 3)) * UHAT_B_STRIDE;

  for (int s = 0; s < 32; ++s) {
    const int nt = (nh << 5) + s;        // N-tile 0..63
    const int n0 = nt << 4;
    const int k  = n0 >> 5;              // upper-capsule index (tile fits in one k)
    const int l0 = n0 & 31;
    const float* wk = wb + k * (D_L * D_H) + l0 + lo16;  // B[j][n] = W[i,k,j,l]
    v8f c = {};
#pragma unroll
    for (int ks = 0; ks < 4; ++ks) {
      const int j = (ks << 2) + khi;
      v2f bv;
      bv.x = wk[j * D_H];
      bv.y = wk[(j + 1) * D_H];
      v2f av = (ks == 0) ? a0 : (ks == 1) ? a1 : (ks == 2) ? a2 : a3;
      // D = A(16x4) * B(4x16) + C ; 8 args: neg_a, A, neg_b, B, c_mod, C, reuse_a, reuse_b
      c = __builtin_amdgcn_wmma_f32_16x16x4_f32(false, av, false, bv, (short)0, c,
                                                false, false);
    }
    // D layout: lane N = lane%16, VGPR v -> M = v + 8*(lane/16)
    float* op = Uhat + rowb + (size_t)i * KL + n0 + lo16;
#pragma unroll
    for (int v = 0; v < 8; ++v) op[(size_t)v * UHAT_B_STRIDE] = c[v];
  }
}

// ---------------------------------------------------------------------------
// Kernel 2: S[b,k,l] = sum_i U_hat[b,i,k,l].  64 blocks x 256 threads, float4.
// ---------------------------------------------------------------------------
__global__ __launch_bounds__(256) void k_reduce_S(const float* __restrict__ Uhat,
                                                  float* __restrict__ S) {
  const int t = blockIdx.x * 256 + threadIdx.x;  // 0..16383
  const int b = t >> 8;
  const int q = t & 255;
  const float* p = Uhat + (size_t)b * UHAT_B_STRIDE + (q << 2);
  v4f acc = {};
  for (int i = 0; i < N_L; ++i) acc += *(const v4f*)(p + (size_t)i * KL);
  *(v4f*)(S + (size_t)b * KL + (q << 2)) = acc;
}

// ---------------------------------------------------------------------------
// Kernel 3: per-(b,i) agreement dot + softmax over k + weighted accumulate,
// then squash.  64 blocks (one per batch) x 256 threads (8 waves).
// Wave w handles i = w, w+8, ...  Lane = upper capsule k (n_h == 32 == wave32).
// LDS rows padded to 33 floats -> conflict-free row reads (33 odd).
// ---------------------------------------------------------------------------
__global__ __launch_bounds__(256) void k_route_squash(const float* __restrict__ Uhat,
                                                      const float* __restrict__ S,
                                                      float* __restrict__ out) {
  __shared__ float s_s[32 * 33];
  __shared__ float s_stage[8][32 * 33];
  __shared__ float s_res[KL];
  __shared__ float s_fac[32];
  const int b    = blockIdx.x;
  const int tid  = threadIdx.x;
  const int wave = tid >> 5;
  const int lane = tid & 31;

#pragma unroll
  for (int r = 0; r < 4; ++r) {
    const int idx = tid + (r << 8);
    s_s[(idx >> 5) * 33 + (idx & 31)] = S[(size_t)b * KL + idx];
  }
  __syncthreads();

  float acc[32];
#pragma unroll
  for (int l = 0; l < 32; ++l) acc[l] = 0.0f;

  const float* ub = Uhat + (size_t)b * UHAT_B_STRIDE;
  float* st = &s_stage[wave][0];

  for (int it = 0; it < 64; ++it) {
    const int i = (it << 3) + wave;
    const float* src = ub + (size_t)i * KL;
#pragma unroll
    for (int t = 0; t < 32; ++t) st[t * 33 + lane] = src[(t << 5) + lane];
    __syncthreads();

    // A_sum[b,i,k]/sqrt(d_h) for k = lane
    float a = 0.0f;
#pragma unroll
    for (int l = 0; l < 32; ++l) a = fmaf(st[lane * 33 + l], s_s[lane * 33 + l], a);
    a *= 0.17677669529663687f;  // 1/sqrt(32)

    // softmax across the 32 lanes (k axis)
    float m = a;
#pragma unroll
    for (int off = 16; off > 0; off >>= 1) m = fmaxf(m, __shfl_xor(m, off, 32));
    const float e = __expf(a - m);
    float ssum = e;
#pragma unroll
    for (int off = 16; off > 0; off >>= 1) ssum += __shfl_xor(ssum, off, 32);
    const float c = e / ssum;

#pragma unroll
    for (int l = 0; l < 32; ++l) acc[l] = fmaf(c, st[lane * 33 + l], acc[l]);
    __syncthreads();
  }

  // cross-wave reduction: each wave parks its accumulator in its own stage
#pragma unroll
  for (int l = 0; l < 32; ++l) st[lane * 33 + l] = acc[l];
  __syncthreads();
#pragma unroll
  for (int r = 0; r < 4; ++r) {
    const int idx = tid + (r << 8);
    const int k = idx >> 5, l = idx & 31;
    float v = 0.0f;
#pragma unroll
    for (int w = 0; w < 8; ++w) v += s_stage[w][k * 33 + l];
    s_res[idx] = v;
  }
  __syncthreads();

  // squash: factor_k = (1 - 1/(exp(||x||)+eps)) / (||x||+eps)
  if (tid < 32) {
    float n2 = 0.0f;
#pragma unroll
    for (int l = 0; l < 32; ++l) { const float x = s_res[tid * 32 + l]; n2 = fmaf(x, x, n2); }
    const float n = sqrtf(n2);
    s_fac[tid] = (1.0f - 1.0f / (__expf(n) + 1e-20f)) / (n + 1e-20f);
  }
  __syncthreads();
#pragma unroll
  for (int r = 0; r < 4; ++r) {
    const int idx = tid + (r << 8);
    out[(size_t)b * KL + idx] = s_res[idx] * s_fac[idx >> 5];
  }
}

// ---------------------------------------------------------------------------
extern "C" void kernel_launch(void* const* d_in, const int* in_sizes, int n_in,
                              void* d_out, int out_size, void* d_ws, size_t ws_size,
                              hipStream_t stream) {
  const float* Ul = (const float*)d_in[0];  // (64, 512, 16)
  const float* W  = (const float*)d_in[1];  // (512, 32, 16, 32)
  float* out = (float*)d_out;               // (64, 32, 32)

  float* Uhat = (float*)d_ws;                                   // 128 MiB
  float* S    = (float*)((char*)d_ws + (size_t)N_B * UHAT_B_STRIDE * sizeof(float));

  k_uhat_wmma<<<dim3(N_L), dim3(256), 0, stream>>>(Ul, W, Uhat);
  k_reduce_S<<<dim3(64), dim3(256), 0, stream>>>(Uhat, S);
  k_route_squash<<<dim3(N_B), dim3(256), 0, stream>>>(Uhat, S, out);
}